// JointNet_14903536517746
// MI455X (gfx1250) — compile-verified
//
#include <hip/hip_runtime.h>

typedef float v2f __attribute__((ext_vector_type(2)));
typedef float v8f __attribute__((ext_vector_type(8)));
typedef float f4  __attribute__((ext_vector_type(4)));

#define ENC_DIM   512
#define DEC_DIM   512
#define VOCAB     1024
#define OUT_DIM   1024   // W row length

// ---------------------------------------------------------------------------
// Kernel 1: out[M, VOCAB] = X[M, 512] * W[:, k_off:k_off+512]^T  (+ bias)
// One wave32 computes one 16x16 tile via V_WMMA_F32_16X16X4_F32 (fp32 path).
// Block = 256 threads = 8 waves arranged 2 (M-tiles) x 4 (N-tiles).
//
// fp32 A 16x4 layout (ISA 7.12.2): lanes 0-15 hold M=lane,K={0,1} in v[0],v[1];
// lanes 16-31 hold M=lane-16, K={2,3}. B 4x16 mirrors with N across lanes.
// C/D 16x16: VGPR j -> (M = j + 8*hi, N = lane&15).
// ---------------------------------------------------------------------------
template <bool ADD_BIAS>
__global__ __launch_bounds__(256)
void proj_gemm_wmma(const float* __restrict__ X,
                    const float* __restrict__ W,
                    const float* __restrict__ bias,
                    float* __restrict__ out,
                    int k_off)
{
    const int wave = threadIdx.x >> 5;
    const int lane = threadIdx.x & 31;
    const int lr   = lane & 15;
    const int hi   = lane >> 4;            // 0: K pair {0,1}, 1: K pair {2,3}

    const int m0 = (blockIdx.y * 2 + (wave >> 2)) * 16;
    const int n0 = (blockIdx.x * 4 + (wave & 3)) * 16;

    const float* __restrict__ arow = X + (size_t)(m0 + lr) * ENC_DIM;           // A row (M = m0+lr)
    const float* __restrict__ brow = W + (size_t)(n0 + lr) * OUT_DIM + k_off;   // B col = W row (N = n0+lr)

    v8f acc = {};

#pragma unroll 8
    for (int k = 0; k < ENC_DIM; k += 4) {
        v2f a = *(const v2f*)(arow + k + hi * 2);   // K = k+2*hi, k+2*hi+1
        v2f b = *(const v2f*)(brow + k + hi * 2);
        acc = __builtin_amdgcn_wmma_f32_16x16x4_f32(
                  /*neg_a=*/false, a, /*neg_b=*/false, b,
                  /*c_mod=*/(short)0, acc, /*reuse_a=*/false, /*reuse_b=*/false);
    }

    const int   col = n0 + lr;
    const float bb  = ADD_BIAS ? bias[col] : 0.0f;
    float* __restrict__ ocol = out + col;
#pragma unroll
    for (int j = 0; j < 8; ++j) {
        const int row = m0 + j + hi * 8;
        ocol[(size_t)row * VOCAB] = acc[j] + bb;
    }
}

// ---------------------------------------------------------------------------
// Kernel 2: log_softmax(enc_proj[bt, :] + dec_proj[b*64+u, :]) -> out[r, :]
// One wave32 per output row of 1024; each lane caches 32 values in registers.
// 8 waves/block cover 8 consecutive u for the same (b,t) -> enc row reuse.
// Output is write-once: non-temporal stores keep proj arrays hot in L2.
// ---------------------------------------------------------------------------
__global__ __launch_bounds__(256)
void joint_logsoftmax(const float* __restrict__ encP,   // (1024, 1024)
                      const float* __restrict__ decP,   // (256, 1024), bias folded in
                      float* __restrict__ out)          // (65536, 1024)
{
    const int  wave = threadIdx.x >> 5;
    const int  lane = threadIdx.x & 31;
    const long long r = (long long)blockIdx.x * 8 + wave;   // 0 .. 65535
    const int  u  = (int)(r & 63);
    const int  bt = (int)(r >> 6);                          // b*256 + t
    const int  b  = bt >> 8;

    const float* __restrict__ e = encP + (size_t)bt * VOCAB;
    const float* __restrict__ d = decP + (size_t)(b * 64 + u) * VOCAB;

    f4 vals[8];
    float mx = -INFINITY;
#pragma unroll
    for (int i = 0; i < 8; ++i) {
        const int v = i * 128 + lane * 4;
        f4 ev = *(const f4*)(e + v);
        f4 dv = *(const f4*)(d + v);
        f4 s  = ev + dv;
        vals[i] = s;
        mx = fmaxf(mx, fmaxf(fmaxf(s.x, s.y), fmaxf(s.z, s.w)));
    }
#pragma unroll
    for (int m = 16; m >= 1; m >>= 1)
        mx = fmaxf(mx, __shfl_xor(mx, m, 32));

    float sum = 0.0f;
#pragma unroll
    for (int i = 0; i < 8; ++i) {
        f4 s = vals[i];
        sum += __expf(s.x - mx) + __expf(s.y - mx) +
               __expf(s.z - mx) + __expf(s.w - mx);
    }
#pragma unroll
    for (int m = 16; m >= 1; m >>= 1)
        sum += __shfl_xor(sum, m, 32);

    const float lse = mx + __logf(sum);

    float* __restrict__ o = out + r * VOCAB;
#pragma unroll
    for (int i = 0; i < 8; ++i) {
        const int v = i * 128 + lane * 4;
        __builtin_nontemporal_store(vals[i] - lse, (f4*)(o + v));
    }
}

// ---------------------------------------------------------------------------
// Launch.
// Inputs: d_in[0] encoder_output (4,256,512) f32
//         d_in[1] decoder_output (4,64,512)  f32
//         d_in[2] W (1024,1024) f32
//         d_in[3] b (1024,)     f32
// Output: (4,256,64,1024) f32
// Workspace: enc_proj 4 MB @ 0, dec_proj 1 MB @ 4 MB.
// ---------------------------------------------------------------------------
extern "C" void kernel_launch(void* const* d_in, const int* in_sizes, int n_in,
                              void* d_out, int out_size, void* d_ws, size_t ws_size,
                              hipStream_t stream)
{
    const float* enc  = (const float*)d_in[0];
    const float* dec  = (const float*)d_in[1];
    const float* W    = (const float*)d_in[2];
    const float* bias = (const float*)d_in[3];
    float*       out  = (float*)d_out;

    float* encP = (float*)d_ws;                                     // 1024*1024 f32
    float* decP = (float*)((char*)d_ws + (size_t)1024 * 1024 * 4);  //  256*1024 f32

    // enc_proj: M=1024 rows -> grid (1024/64 N-blocks = 16, 1024/32 M-blocks = 32)
    proj_gemm_wmma<false><<<dim3(16, 32), 256, 0, stream>>>(enc, W, bias, encP, 0);
    // dec_proj (+bias): M=256 rows, W columns 512..1023
    proj_gemm_wmma<true ><<<dim3(16, 8), 256, 0, stream>>>(dec, W, bias, decP, ENC_DIM);

    // 65536 rows / 8 waves per block
    joint_logsoftmax<<<8192, 256, 0, stream>>>(encP, decP, out);
}